// RegressLoss_GPU_72670846648745
// MI455X (gfx1250) — compile-verified
//
#include <hip/hip_runtime.h>
#include <hip/hip_bf16.h>

// ---------------------------------------------------------------------------
// result = sum_j W[j] * (sum_i |yt[i,j]-yp[i,j]|) / (sum_i yt[i,j])
// Memory-bound: 167.8 MB streamed once -> ~7.2us floor at 23.3 TB/s.
// Stage 1: b128 nontemporal streaming loads, 4 rows (=5 float4) per thread
//          per iteration, deterministic shfl+LDS tree, per-block partials.
// Stage 2: single wave32; cross-block reduction finished with one
//          v_wmma_f32_16x16x4_f32 (A = 16x4 partial sums, B = ones).
// ---------------------------------------------------------------------------

typedef float  f4  __attribute__((ext_vector_type(4)));
typedef float  v2f __attribute__((ext_vector_type(2)));
typedef float  v8f __attribute__((ext_vector_type(8)));

#define THREADS 256
#define BLOCKS  1024
#define WAVES   (THREADS / 32)

__global__ __launch_bounds__(THREADS)
void regress_stage1(const float* __restrict__ yt,
                    const float* __restrict__ yp,
                    float* __restrict__ partials,   // [BLOCKS][10]
                    long long numGroups,            // N / 4 (groups of 4 rows)
                    long long N)
{
    float sT[5] = {0.f, 0.f, 0.f, 0.f, 0.f};   // column sums of y_true
    float sA[5] = {0.f, 0.f, 0.f, 0.f, 0.f};   // column sums of |yt-yp|

    const long long stride = (long long)gridDim.x * blockDim.x;
    const long long tid0   = (long long)blockIdx.x * blockDim.x + threadIdx.x;

    for (long long g = tid0; g < numGroups; g += stride) {
        // 4 rows = 20 floats = 80 bytes: 16B-aligned for every g.
        const f4* t4 = (const f4*)(yt + g * 20);
        const f4* p4 = (const f4*)(yp + g * 20);
        float t[20], p[20];
#pragma unroll
        for (int i = 0; i < 5; ++i) {
            f4 a = __builtin_nontemporal_load(&t4[i]);   // global_load_b128, TH=NT
            f4 b = __builtin_nontemporal_load(&p4[i]);
            t[4*i+0] = a.x; t[4*i+1] = a.y; t[4*i+2] = a.z; t[4*i+3] = a.w;
            p[4*i+0] = b.x; p[4*i+1] = b.y; p[4*i+2] = b.z; p[4*i+3] = b.w;
        }
#pragma unroll
        for (int k = 0; k < 20; ++k) {
            const int c = k % 5;                 // compile-time after unroll
            sT[c] += t[k];
            sA[c] += fabsf(t[k] - p[k]);
        }
    }

    // Tail rows (N % 4) handled by one thread, deterministic.
    if (tid0 == 0) {
        for (long long r = numGroups * 4; r < N; ++r) {
#pragma unroll
            for (int c = 0; c < 5; ++c) {
                const float a = yt[r*5 + c], b = yp[r*5 + c];
                sT[c] += a;
                sA[c] += fabsf(a - b);
            }
        }
    }

    // Wave32 tree reduction (deterministic).
#pragma unroll
    for (int off = 16; off > 0; off >>= 1) {
#pragma unroll
        for (int c = 0; c < 5; ++c) {
            sT[c] += __shfl_down(sT[c], (unsigned)off, 32);
            sA[c] += __shfl_down(sA[c], (unsigned)off, 32);
        }
    }

    __shared__ float red[WAVES * 10];
    const int lane = threadIdx.x & 31;
    const int wave = threadIdx.x >> 5;
    if (lane == 0) {
#pragma unroll
        for (int c = 0; c < 5; ++c) {
            red[wave*10 + c]     = sT[c];
            red[wave*10 + 5 + c] = sA[c];
        }
    }
    __syncthreads();

    if (threadIdx.x < 10) {
        float v = 0.f;
#pragma unroll
        for (int w = 0; w < WAVES; ++w) v += red[w*10 + threadIdx.x];
        partials[(long long)blockIdx.x * 10 + threadIdx.x] = v;
    }
}

// One wave32. Finishes the cross-block reduction with a single
// V_WMMA_F32_16X16X4_F32: A[m][k] = partial sums of quantity m over block
// group k (blocks interleaved mod 4); B = all-ones 4x16 (layout-free);
// D[m][0] = total[m]. A layout per ISA: lanes 0-15 hold K=0,1 (VGPR0,1),
// lanes 16-31 hold K=2,3. D layout: VGPR v = row M (lane<16) / M+8, N=lane%16.
__global__ __launch_bounds__(32)
void regress_stage2(const float* __restrict__ partials,
                    float* __restrict__ out,
                    int numBlocks)
{
    const int L  = threadIdx.x;
    const int m  = L & 15;          // A-matrix row this lane feeds
    const int g0 = (L >> 4) * 2;    // first K-slot owned by this lane half

    float a0 = 0.f, a1 = 0.f;
    if (m < 10) {                   // rows 10..15 stay zero
        for (int b = g0;     b < numBlocks; b += 4) a0 += partials[b*10 + m];
        for (int b = g0 + 1; b < numBlocks; b += 4) a1 += partials[b*10 + m];
    }
    // EXEC is all-1s again here (divergence closed) -> WMMA legal.
    v2f A; A[0] = a0;  A[1] = a1;
    v2f B; B[0] = 1.f; B[1] = 1.f;
    v8f C = {};
    v8f D = __builtin_amdgcn_wmma_f32_16x16x4_f32(
        /*neg_a=*/false, A, /*neg_b=*/false, B,
        /*c_mod=*/(short)0, C, /*reuse_a=*/false, /*reuse_b=*/false);

    __shared__ float tot[16];
    if (L == 0) {                   // N=0 column, rows 0..7
#pragma unroll
        for (int v = 0; v < 8; ++v) tot[v] = D[v];
    }
    if (L == 16) {                  // N=0 column, rows 8..15
#pragma unroll
        for (int v = 0; v < 8; ++v) tot[8 + v] = D[v];
    }
    __syncthreads();

    if (L == 0) {
        const float W[5] = {0.3f, 0.175f, 0.175f, 0.175f, 0.175f};
        float r = 0.f;
#pragma unroll
        for (int j = 0; j < 5; ++j) r += W[j] * tot[5 + j] / tot[j];
        out[0] = r;                 // the /N factors cancel exactly
    }
}

extern "C" void kernel_launch(void* const* d_in, const int* in_sizes, int n_in,
                              void* d_out, int out_size, void* d_ws, size_t ws_size,
                              hipStream_t stream)
{
    const float* yt = (const float*)d_in[0];
    const float* yp = (const float*)d_in[1];
    float* partials = (float*)d_ws;             // BLOCKS*10 floats = 40 KB

    const long long total     = (long long)in_sizes[0];  // N*5
    const long long N         = total / 5;
    const long long numGroups = N / 4;

    regress_stage1<<<BLOCKS, THREADS, 0, stream>>>(yt, yp, partials, numGroups, N);
    regress_stage2<<<1, 32, 0, stream>>>(partials, (float*)d_out, BLOCKS);
}